// GraphTransformerBlock_18322330484760
// MI455X (gfx1250) — compile-verified
//
#include <hip/hip_runtime.h>
#include <hip/hip_bf16.h>

// ---------------------------------------------------------------------------
// Graph Transformer block for MI455X (gfx1250), fp32 end-to-end.
// - All matmuls: V_WMMA_F32_16X16X4_F32 (fp32 A/B, fp32 acc) -> reference
//   precision for the discrete top-K neighbor selection.
// - GEMM waves own a 16x64 C strip (1 A fragment feeds 4 accumulators).
// - Edge-feature x tiles staged into LDS with GLOBAL_LOAD_ASYNC_TO_LDS_B128
//   (ASYNCcnt-tracked, VGPR-bypassing) instead of register round-trips.
// - The (N,N,D) pairwise-diff tensor (604 MB in the reference) is never
//   materialized: tiled 16x16 with LDS-staged x slices.
// ---------------------------------------------------------------------------

static constexpr int N_TOK  = 768;   // N
static constexpr int DMODEL = 256;   // D
static constexpr int NHEAD  = 4;     // H
static constexpr int HD     = 64;    // head dim
static constexpr int TOPK   = 16;    // K
static constexpr int DFF    = 512;   // 2*D
static constexpr int ADJW   = N_TOK / 32;   // adjacency bitmask words per row
static constexpr int PSTR   = N_TOK + 4;    // padded LDS stride (softmax strip)
static constexpr int XSTR   = DMODEL + 4;   // padded LDS stride, 16B-aligned rows

typedef float v2f __attribute__((ext_vector_type(2)));
typedef float v8f __attribute__((ext_vector_type(8)));
typedef int   v4i __attribute__((ext_vector_type(4)));

static __device__ __forceinline__ v8f wmma_f32(v2f a, v2f b, v8f c) {
#if __has_builtin(__builtin_amdgcn_wmma_f32_16x16x4_f32)
  return __builtin_amdgcn_wmma_f32_16x16x4_f32(false, a, false, b, (short)0, c,
                                               false, false);
#else
  return c;  // compile-safety fallback; never expected on gfx1250
#endif
}

// Async global->LDS 16-byte copy (tracked by ASYNCcnt).
// Probe-derived signature: (v4i AS1*, v4i AS3*, imm offset, imm cpol).
static __device__ __forceinline__ void async_copy_b128(const float* g, float* l) {
#if __has_builtin(__builtin_amdgcn_global_load_async_to_lds_b128)
  __builtin_amdgcn_global_load_async_to_lds_b128(
      (__attribute__((address_space(1))) v4i*)g,
      (__attribute__((address_space(3))) v4i*)l, 0, 0);
#else
  unsigned dst = (unsigned)(uintptr_t)(__attribute__((address_space(3))) void*)l;
  asm volatile("global_load_async_to_lds_b128 %0, %1, off"
               :: "v"(dst), "v"(g) : "memory");
#endif
}

static __device__ __forceinline__ void wait_async0() {
#if __has_builtin(__builtin_amdgcn_s_wait_asynccnt)
  __builtin_amdgcn_s_wait_asynccnt(0);
#else
  asm volatile("s_wait_asynccnt 0x0" ::: "memory");
#endif
}

// ---------------------------------------------------------------------------
// 1) Row L2-normalize: dn[n] = x[n] / max(||x[n]||, 1e-12)
// ---------------------------------------------------------------------------
__global__ __launch_bounds__(256) void rownorm_kernel(const float* __restrict__ x,
                                                      float* __restrict__ dn) {
  __shared__ float red[256];
  int row = blockIdx.x, tid = threadIdx.x;
  float v = x[row * DMODEL + tid];
  red[tid] = v * v;
  __syncthreads();
  for (int s = 128; s > 0; s >>= 1) {
    if (tid < s) red[tid] += red[tid + s];
    __syncthreads();
  }
  float inv = 1.0f / fmaxf(sqrtf(red[0]), 1e-12f);
  dn[row * DMODEL + tid] = v * inv;
}

// ---------------------------------------------------------------------------
// fp32 WMMA GEMM: one wave owns a 16x(16*NT) C strip; 1 A fragment feeds NT
// accumulators per k-step (4x less A traffic than 1 tile/wave).
//   EPI 0: C = acc (+ bias[col] if bias)      EPI 1: QKV (H,HD,3) scatter
//   EPI 2: C = acc + bias[col] + resid        (BT: B read transposed)
// Fragment layout per CDNA5 ISA: A 16x4 (lanes 0-15 -> K0/K1, 16-31 -> K2/K3),
// C/D: VGPR i holds (M=i, lanes 0-15) and (M=i+8, lanes 16-31).
// ---------------------------------------------------------------------------
template <int EPI, bool BT, int NT>
__global__ __launch_bounds__(32) void gemm16_kernel(
    const float* __restrict__ A, int lda, const float* __restrict__ B, int ldb,
    const float* __restrict__ bias, const float* __restrict__ resid,
    float* __restrict__ C, int ldc, int Kdim) {
  int row0 = blockIdx.x * 16, col0 = blockIdx.y * 16 * NT;
  int lane = threadIdx.x;
  int hw = lane >> 4;      // which 16-lane half
  int r  = lane & 15;
  v8f acc[NT] = {};
  for (int k0 = 0; k0 < Kdim; k0 += 4) {
    int ka = k0 + 2 * hw;
    v2f a;
    a.x = A[(row0 + r) * lda + ka];
    a.y = A[(row0 + r) * lda + ka + 1];
    if (k0 + 8 < Kdim)  // hide L2 latency on the long K loops
      __builtin_prefetch(&A[(row0 + r) * lda + ka + 8], 0, 1);
#pragma unroll
    for (int t = 0; t < NT; ++t) {
      int cb = col0 + t * 16 + r;
      v2f b;
      if (BT) {  // B(k, col) = Bsrc[col*ldb + k]
        b.x = B[cb * ldb + ka];
        b.y = B[cb * ldb + ka + 1];
      } else {   // B row-major KxN
        b.x = B[ka * ldb + cb];
        b.y = B[(ka + 1) * ldb + cb];
      }
      acc[t] = wmma_f32(a, b, acc[t]);
    }
  }
#pragma unroll
  for (int t = 0; t < NT; ++t) {
#pragma unroll
    for (int i = 0; i < 8; ++i) {
      int row = row0 + i + 8 * hw;
      int col = col0 + t * 16 + r;
      float v = acc[t][i];
      if (EPI == 0) {
        if (bias) v += bias[col];
        C[row * ldc + col] = v;
      } else if (EPI == 1) {  // qkv: flat col j = h*(HD*3) + d*3 + s
        v += bias[col];
        int s = col % 3, d = (col / 3) & (HD - 1), h = col / (3 * HD);
        C[(((s * NHEAD) + h) * N_TOK + row) * HD + d] = v;
      } else {  // EPI == 2
        C[row * ldc + col] = v + bias[col] + resid[row * ldc + col];
      }
    }
  }
}

// ---------------------------------------------------------------------------
// 2) Zero the adjacency bitmask (ws is poisoned; atomicOr accumulates).
// ---------------------------------------------------------------------------
__global__ void zero_u32_kernel(unsigned* __restrict__ p, int n) {
  int i = blockIdx.x * blockDim.x + threadIdx.x;
  if (i < n) p[i] = 0u;
}

// ---------------------------------------------------------------------------
// 3) Per-row top-16 on an LDS copy of sim (keeps sim intact for edge feats).
//    Tie-break to smaller index (matches lax.top_k stability). Adjacency is
//    symmetrized via idempotent atomicOr; diagonal set (self loops).
// ---------------------------------------------------------------------------
__global__ __launch_bounds__(256) void topk_adj_kernel(const float* __restrict__ sim,
                                                       unsigned* __restrict__ adj) {
  __shared__ float srow[N_TOK];
  __shared__ float rv[256];
  __shared__ int   ri[256];
  int row = blockIdx.x, tid = threadIdx.x;
  float ninf = -__builtin_inff();
  for (int m = tid; m < N_TOK; m += 256)
    srow[m] = (m == row) ? ninf : sim[row * N_TOK + m];
  if (tid == 0) atomicOr(&adj[row * ADJW + (row >> 5)], 1u << (row & 31));
  __syncthreads();
  for (int t = 0; t < TOPK; ++t) {
    float bv = ninf;
    int bi = -1;
    for (int m = tid; m < N_TOK; m += 256) {
      float v = srow[m];
      if (v > bv || (v == bv && bi >= 0 && m < bi)) { bv = v; bi = m; }
      if (bi < 0 && v > ninf) { bv = v; bi = m; }
    }
    rv[tid] = bv; ri[tid] = bi;
    __syncthreads();
    for (int s = 128; s > 0; s >>= 1) {
      if (tid < s) {
        float ov = rv[tid + s]; int oi = ri[tid + s];
        if (oi >= 0 && (ov > rv[tid] || (ov == rv[tid] && oi < ri[tid]) || ri[tid] < 0)) {
          rv[tid] = ov; ri[tid] = oi;
        }
      }
      __syncthreads();
    }
    int m = ri[0];
    if (tid == 0 && m >= 0) {
      srow[m] = ninf;
      atomicOr(&adj[row * ADJW + (m >> 5)], 1u << (m & 31));
      atomicOr(&adj[m * ADJW + (row >> 5)], 1u << (row & 31));
    }
    __syncthreads();
  }
}

// ---------------------------------------------------------------------------
// 4) Fused edge features + edge MLP, 16x16 pair tile per workgroup.
//    x tiles staged via async global->LDS B128 (ASYNCcnt), rows padded to
//    260 floats: 16B-aligned rows, and the column-indexed access walks
//    (4*row + d) % 64 banks -> conflict-free.
//    ef = {sim, ||xn-xm||/16, mean|xn-xm|, max|xn-xm|} -> 4->8 relu -> 8->H.
// ---------------------------------------------------------------------------
__global__ __launch_bounds__(256) void edge_bias_kernel(
    const float* __restrict__ x, const float* __restrict__ sim,
    const float* __restrict__ e1w, const float* __restrict__ e1b,
    const float* __restrict__ e2w, const float* __restrict__ e2b,
    const float* __restrict__ gatep, float* __restrict__ gbias) {
  __shared__ float xa[16][XSTR];
  __shared__ float xb[16][XSTR];
  __shared__ float w1[32], b1[8], w2[32], b2s[4];
  int n0 = blockIdx.x * 16, m0 = blockIdx.y * 16;
  int tid = threadIdx.x;
  // async stage: 16 rows * 64 chunks of 16B, per tile -> 4 iter * 2 issues/thread
  for (int i = tid; i < 16 * (DMODEL / 4); i += 256) {
    int r = i >> 6, c4 = (i & 63) * 4;
    async_copy_b128(&x[(n0 + r) * DMODEL + c4], &xa[r][c4]);
    async_copy_b128(&x[(m0 + r) * DMODEL + c4], &xb[r][c4]);
  }
  if (tid < 32) w1[tid] = e1w[tid];
  else if (tid < 40) b1[tid - 32] = e1b[tid - 32];
  else if (tid < 72) w2[tid - 40] = e2w[tid - 40];
  else if (tid < 76) b2s[tid - 72] = e2b[tid - 72];
  wait_async0();        // this wave's async LDS writes complete
  __syncthreads();      // all waves' writes visible
  int tm = tid & 15, tn = tid >> 4;
  float ssq = 0.f, sab = 0.f, mab = 0.f;
#pragma unroll 4
  for (int d = 0; d < DMODEL; ++d) {
    float df = xa[tn][d] - xb[tm][d];
    float ad = fabsf(df);
    ssq = fmaf(df, df, ssq);
    sab += ad;
    mab = fmaxf(mab, ad);
  }
  int n = n0 + tn, m = m0 + tm;
  float ef[4] = {sim[n * N_TOK + m], sqrtf(ssq) * (1.0f / 16.0f),
                 sab * (1.0f / DMODEL), mab};
  float hbuf[8];
#pragma unroll
  for (int j = 0; j < 8; ++j) {
    float s = b1[j];
#pragma unroll
    for (int i = 0; i < 4; ++i) s = fmaf(ef[i], w1[i * 8 + j], s);
    hbuf[j] = fmaxf(s, 0.f);
  }
  float gate = gatep[0];
#pragma unroll
  for (int h = 0; h < NHEAD; ++h) {
    float s = b2s[h];
#pragma unroll
    for (int j = 0; j < 8; ++j) s = fmaf(hbuf[j], w2[j * 4 + h], s);
    gbias[(h * N_TOK + n) * N_TOK + m] = gate * s;
  }
}

// ---------------------------------------------------------------------------
// 5) Fused attention: one wave per (head, 16-row tile). Full 16x768 logit
//    strip lives in ~48 KB LDS (CDNA5: 320 KB/WGP) -> single-pass softmax.
//    qk^T and P*V via f32 WMMA; ctx written in (N, 256) layout.
// ---------------------------------------------------------------------------
__global__ __launch_bounds__(32) void attn_kernel(const float* __restrict__ qkv,
                                                  const float* __restrict__ gbias,
                                                  const unsigned* __restrict__ adj,
                                                  float* __restrict__ ctx) {
  __shared__ float P[16][PSTR];
  int h = blockIdx.y, n0 = blockIdx.x * 16;
  int lane = threadIdx.x, hw = lane >> 4, r = lane & 15;
  const float* qb = qkv + ((0 * NHEAD + h) * N_TOK) * HD;
  const float* kb = qkv + ((1 * NHEAD + h) * N_TOK) * HD;
  const float* vb = qkv + ((2 * NHEAD + h) * N_TOK) * HD;
  float ninf = -__builtin_inff();

  // logits = q k^T / sqrt(HD) + gate*bias, masked by adjacency bitmask
  for (int mt = 0; mt < N_TOK / 16; ++mt) {
    int m0 = mt * 16;
    v8f acc = {};
    for (int kk = 0; kk < HD; kk += 4) {
      int ka = kk + 2 * hw;
      v2f a, b;
      a.x = qb[(n0 + r) * HD + ka];
      a.y = qb[(n0 + r) * HD + ka + 1];
      b.x = kb[(m0 + r) * HD + ka];   // B = k^T
      b.y = kb[(m0 + r) * HD + ka + 1];
      acc = wmma_f32(a, b, acc);
    }
#pragma unroll
    for (int i = 0; i < 8; ++i) {
      int li = i + 8 * hw;
      int n = n0 + li, m = m0 + r;
      float sv = acc[i] * 0.125f + gbias[(h * N_TOK + n) * N_TOK + m];
      unsigned w = adj[n * ADJW + (m >> 5)];
      if (!((w >> (m & 31)) & 1u)) sv = ninf;
      P[li][m] = sv;
    }
  }
  __syncthreads();

  // row softmax (wave-wide reductions, warpSize == 32)
  for (int li = 0; li < 16; ++li) {
    float mx = ninf;
    for (int m = lane; m < N_TOK; m += 32) mx = fmaxf(mx, P[li][m]);
    for (int off = 16; off > 0; off >>= 1) mx = fmaxf(mx, __shfl_xor(mx, off, 32));
    float sum = 0.f;
    for (int m = lane; m < N_TOK; m += 32) {
      float e = expf(P[li][m] - mx);
      P[li][m] = e;
      sum += e;
    }
    for (int off = 16; off > 0; off >>= 1) sum += __shfl_xor(sum, off, 32);
    float inv = 1.0f / sum;
    for (int m = lane; m < N_TOK; m += 32) P[li][m] *= inv;
  }
  __syncthreads();

  // ctx = P (16x768) @ V (768x64); store at ctx[n][h*64 + d]
  for (int ct = 0; ct < HD / 16; ++ct) {
    int c0 = ct * 16;
    v8f acc = {};
    for (int kk = 0; kk < N_TOK; kk += 4) {
      int ka = kk + 2 * hw;
      v2f a, b;
      a.x = P[r][ka];
      a.y = P[r][ka + 1];
      b.x = vb[ka * HD + c0 + r];
      b.y = vb[(ka + 1) * HD + c0 + r];
      acc = wmma_f32(a, b, acc);
    }
#pragma unroll
    for (int i = 0; i < 8; ++i) {
      int li = i + 8 * hw;
      ctx[(n0 + li) * DMODEL + h * HD + c0 + r] = acc[i];
    }
  }
}

// ---------------------------------------------------------------------------
// 6) concat [x, msg] -> (N, 512)
// ---------------------------------------------------------------------------
__global__ void concat_kernel(const float* __restrict__ x,
                              const float* __restrict__ msg,
                              float* __restrict__ cat) {
  int i = blockIdx.x * blockDim.x + threadIdx.x;
  if (i >= N_TOK * DFF) return;
  int row = i / DFF, c = i % DFF;
  cat[i] = (c < DMODEL) ? x[row * DMODEL + c] : msg[row * DMODEL + (c - DMODEL)];
}

// ---------------------------------------------------------------------------
// 7) LayerNorm (two-pass, matches reference) + exact GELU, in place on (N,512)
// ---------------------------------------------------------------------------
__global__ __launch_bounds__(256) void ln_gelu_kernel(float* __restrict__ y,
                                                      const float* __restrict__ g,
                                                      const float* __restrict__ b) {
  __shared__ float red[256];
  int row = blockIdx.x, tid = threadIdx.x;
  float a = y[row * DFF + tid];
  float c = y[row * DFF + tid + 256];
  red[tid] = a + c;
  __syncthreads();
  for (int s = 128; s > 0; s >>= 1) { if (tid < s) red[tid] += red[tid + s]; __syncthreads(); }
  float mu = red[0] * (1.0f / DFF);
  __syncthreads();
  float da = a - mu, dc = c - mu;
  red[tid] = da * da + dc * dc;
  __syncthreads();
  for (int s = 128; s > 0; s >>= 1) { if (tid < s) red[tid] += red[tid + s]; __syncthreads(); }
  float inv = rsqrtf(red[0] * (1.0f / DFF) + 1e-5f);
  const float kInvSqrt2 = 0.70710678118654752f;
  float t = da * inv * g[tid] + b[tid];
  t = 0.5f * t * (1.0f + erff(t * kInvSqrt2));
  y[row * DFF + tid] = t;
  float t2 = dc * inv * g[tid + 256] + b[tid + 256];
  t2 = 0.5f * t2 * (1.0f + erff(t2 * kInvSqrt2));
  y[row * DFF + tid + 256] = t2;
}

// ---------------------------------------------------------------------------
// launcher
// ---------------------------------------------------------------------------
extern "C" void kernel_launch(void* const* d_in, const int* in_sizes, int n_in,
                              void* d_out, int out_size, void* d_ws, size_t ws_size,
                              hipStream_t stream) {
  (void)in_sizes; (void)n_in; (void)out_size; (void)ws_size;
  const float* x       = (const float*)d_in[0];
  const float* Wqkv_w  = (const float*)d_in[1];
  const float* Wqkv_b  = (const float*)d_in[2];
  const float* out_w   = (const float*)d_in[3];
  const float* out_b   = (const float*)d_in[4];
  const float* ffn1_w  = (const float*)d_in[5];
  const float* ffn1_b  = (const float*)d_in[6];
  const float* ln_g    = (const float*)d_in[7];
  const float* ln_b    = (const float*)d_in[8];
  const float* ffn2_w  = (const float*)d_in[9];
  const float* ffn2_b  = (const float*)d_in[10];
  const float* e1_w    = (const float*)d_in[11];
  const float* e1_b    = (const float*)d_in[12];
  const float* e2_w    = (const float*)d_in[13];
  const float* e2_b    = (const float*)d_in[14];
  const float* gate    = (const float*)d_in[15];

  // workspace layout (float units) — total ≈ 18.9 MB (L2-resident: 192 MB)
  float* ws = (float*)d_ws;
  float* dn    = ws;                               // 768*256
  float* simb  = dn   + N_TOK * DMODEL;            // 768*768
  float* qkvb  = simb + N_TOK * N_TOK;             // 3*4*768*64
  float* ctx   = qkvb + 3 * NHEAD * N_TOK * HD;    // 768*256
  float* msg   = ctx  + N_TOK * DMODEL;            // 768*256
  float* cat   = msg  + N_TOK * DMODEL;            // 768*512
  float* y1    = cat  + N_TOK * DFF;               // 768*512
  float* gbias = y1   + N_TOK * DFF;               // 4*768*768
  unsigned* adj = (unsigned*)(gbias + NHEAD * N_TOK * N_TOK);  // 768*24 words
  float* out   = (float*)d_out;

  const dim3 b32(32), b256(256);
  const int MT = N_TOK / 16;  // 48 row tiles

  // 1) normalize rows (forward: stop_gradient is identity -> one sim matrix)
  rownorm_kernel<<<N_TOK, b256, 0, stream>>>(x, dn);
  // 2) sim = dn dn^T   (f32 WMMA, B transposed-read, 16x64 strip/wave)
  gemm16_kernel<0, true, 4><<<dim3(MT, N_TOK / 64), b32, 0, stream>>>(
      dn, DMODEL, dn, DMODEL, nullptr, nullptr, simb, N_TOK, DMODEL);
  // 3) adjacency: zero, then top-16 + symmetrize + self loops
  zero_u32_kernel<<<(N_TOK * ADJW + 255) / 256, b256, 0, stream>>>(adj, N_TOK * ADJW);
  topk_adj_kernel<<<N_TOK, b256, 0, stream>>>(simb, adj);
  // 4) qkv = x @ Wqkv + b, scattered to (s, h, n, d)
  gemm16_kernel<1, false, 4><<<dim3(MT, (3 * DMODEL) / 64), b32, 0, stream>>>(
      x, DMODEL, Wqkv_w, 3 * DMODEL, Wqkv_b, nullptr, qkvb, 0, DMODEL);
  // 5) gated edge bias (H, N, N); async-LDS staged, diff never materialized
  edge_bias_kernel<<<dim3(MT, MT), b256, 0, stream>>>(
      x, simb, e1_w, e1_b, e2_w, e2_b, gate, gbias);
  // 6) fused masked attention -> ctx (N, 256)
  attn_kernel<<<dim3(MT, NHEAD), b32, 0, stream>>>(qkvb, gbias, adj, ctx);
  // 7) msg = ctx @ out_w + out_b
  gemm16_kernel<0, false, 4><<<dim3(MT, DMODEL / 64), b32, 0, stream>>>(
      ctx, DMODEL, out_w, DMODEL, out_b, nullptr, msg, DMODEL, DMODEL);
  // 8) cat = [x, msg]
  concat_kernel<<<(N_TOK * DFF + 255) / 256, b256, 0, stream>>>(x, msg, cat);
  // 9) y1 = cat @ ffn1_w + b
  gemm16_kernel<0, false, 4><<<dim3(MT, DFF / 64), b32, 0, stream>>>(
      cat, DFF, ffn1_w, DFF, ffn1_b, nullptr, y1, DFF, DFF);
  // 10) LayerNorm + exact GELU in place
  ln_gelu_kernel<<<N_TOK, b256, 0, stream>>>(y1, ln_g, ln_b);
  // 11) out = x + y1 @ ffn2_w + b   (fused residual epilogue)
  gemm16_kernel<2, false, 4><<<dim3(MT, DMODEL / 64), b32, 0, stream>>>(
      y1, DFF, ffn2_w, DMODEL, ffn2_b, x, out, DMODEL, DFF);
}